// CRF_6055903887764
// MI455X (gfx1250) — compile-verified
//
#include <hip/hip_runtime.h>
#include <hip/hip_bf16.h>

// CRF forward-algorithm NLL on gfx1250 (MI455X).
// Strategy: associative log-matmul scan -> per-chunk exp-space 16x16 matrix
// products on V_WMMA_F32_16X16X4_F32 (4 chained WMMAs per step, K=16),
// with exact power-of-two rescaling; single-wave log-space fold of the 128
// chunk matrices; gold path score as a parallel reduction.

typedef float v2f __attribute__((ext_vector_type(2)));
typedef float v8f __attribute__((ext_vector_type(8)));

#define TNUM    16
#define SEQL    8192
#define NCHUNK  128
#define CHLEN   64                    // NCHUNK * CHLEN == SEQL
#define MAT_OFF   0                   // NCHUNK * 256 floats (chunk matrices, m-major)
#define SCALE_OFF (NCHUNK * 256)      // NCHUNK floats (power-of-two exponents)
#define GOLD_OFF  (NCHUNK * 256 + NCHUNK) // NCHUNK floats (gold partial sums)

// ds_swizzle group-of-32: offset = xor<<10 | or<<5 | and ; and=0x1f, or=0
#define SWZF(v, imm) (__int_as_float(__builtin_amdgcn_ds_swizzle(__float_as_int(v), (imm))))
#define SWZ_X1  0x041F
#define SWZ_X2  0x081F
#define SWZ_X4  0x101F
#define SWZ_X8  0x201F
#define SWZ_X16 0x401F

__global__ __launch_bounds__(32) void crf_phase1(
    const float* __restrict__ emit,     // [V, 16]
    const float* __restrict__ trans,    // [16, 16]
    const int*   __restrict__ x,        // [L]
    const int*   __restrict__ y,        // [L]
    float*       __restrict__ ws)
{
    const int c    = blockIdx.x;
    const int lane = threadIdx.x & 31;
    const int hi   = lane >> 4;         // 0: lanes 0-15, 1: lanes 16-31
    const int m    = lane & 15;
    const int t0   = c * CHLEN;
    const int t1   = t0 + CHLEN;

    // Constant A operand: Texp^T = exp(trans)^T in WMMA A-layout.
    // A-layout (m,k): lane = m + 16*(k%4>=2), reg = (k%4)&1, block kb = k/4.
    v2f aT[4];
    #pragma unroll
    for (int kb = 0; kb < 4; ++kb) {
        const int k0 = 4 * kb + (hi ? 2 : 0);
        aT[kb].x = __expf(trans[(k0    ) * TNUM + m]);
        aT[kb].y = __expf(trans[(k0 + 1) * TNUM + m]);
    }

    // D = Pe^T starts as identity, in WMMA C/D layout:
    // reg v, lane l -> row (v + 8*hi), col (l&15).
    v8f d;
    #pragma unroll
    for (int v = 0; v < 8; ++v) d[v] = ((v + 8 * hi) == m) ? 1.0f : 0.0f;
    int chunk_ex = 0;

    const bool lo = (hi == 0);
    float w = __expf(emit[x[t0] * TNUM + m]);   // w_t[j] at lane j (and j+16)

    for (int t = t0; t < t1; ++t) {
        // Prefetch next step's emission weight (independent of the chain).
        float wn = 0.0f;
        if (t + 1 < t1) wn = __expf(emit[x[t + 1] * TNUM + m]);

        // A_t = diag(w_t) * Texp^T : row scale == per-lane multiply.
        v2f A0 = aT[0] * w, A1 = aT[1] * w, A2 = aT[2] * w, A3 = aT[3] * w;

        // Build B blocks (rows k of Pe^T) from D's C-layout:
        // half-wave swaps + selects.
        float sw0 = SWZF(d[0], SWZ_X16), sw1 = SWZF(d[1], SWZ_X16);
        float sw2 = SWZF(d[2], SWZ_X16), sw3 = SWZF(d[3], SWZ_X16);
        float sw4 = SWZF(d[4], SWZ_X16), sw5 = SWZF(d[5], SWZ_X16);
        float sw6 = SWZF(d[6], SWZ_X16), sw7 = SWZF(d[7], SWZ_X16);
        v2f B0, B1, B2, B3;
        B0.x = lo ? d[0] : sw2;  B0.y = lo ? d[1] : sw3;   // rows 0,2 | 1,3
        B1.x = lo ? d[4] : sw6;  B1.y = lo ? d[5] : sw7;   // rows 4,6 | 5,7
        B2.x = lo ? sw0 : d[2];  B2.y = lo ? sw1 : d[3];   // rows 8,10 | 9,11
        B3.x = lo ? sw4 : d[6];  B3.y = lo ? sw5 : d[7];   // rows 12,14 | 13,15

        // Pe^T <- A_t x Pe^T  (16x16x16 via 4 chained 16x16x4 WMMAs)
        v8f acc;
        #pragma unroll
        for (int v = 0; v < 8; ++v) acc[v] = 0.0f;
        acc = __builtin_amdgcn_wmma_f32_16x16x4_f32(false, A0, false, B0, (short)0, acc, false, false);
        acc = __builtin_amdgcn_wmma_f32_16x16x4_f32(false, A1, false, B1, (short)0, acc, false, false);
        acc = __builtin_amdgcn_wmma_f32_16x16x4_f32(false, A2, false, B2, (short)0, acc, false, false);
        acc = __builtin_amdgcn_wmma_f32_16x16x4_f32(false, A3, false, B3, (short)0, acc, false, false);
        d = acc;

        // Exact power-of-two rescale every 4 steps (worst growth ~2^62 < 2^127).
        if (((t - t0) & 3) == 3) {
            float lm = fmaxf(fmaxf(fmaxf(d[0], d[1]), fmaxf(d[2], d[3])),
                             fmaxf(fmaxf(d[4], d[5]), fmaxf(d[6], d[7])));
            lm = fmaxf(lm, SWZF(lm, SWZ_X1));
            lm = fmaxf(lm, SWZF(lm, SWZ_X2));
            lm = fmaxf(lm, SWZF(lm, SWZ_X4));
            lm = fmaxf(lm, SWZF(lm, SWZ_X8));
            lm = fmaxf(lm, SWZF(lm, SWZ_X16));
            const int ex = ((__float_as_int(lm) >> 23) & 255) - 126;
            const float s = ldexpf(1.0f, -ex);
            #pragma unroll
            for (int v = 0; v < 8; ++v) d[v] *= s;
            chunk_ex += ex;
        }
        w = wn;
    }

    // Store chunk matrix row-major: ws[c*256 + row*16 + col].
    #pragma unroll
    for (int v = 0; v < 8; ++v)
        ws[MAT_OFF + c * 256 + (v + 8 * hi) * TNUM + m] = d[v];
    if (lane == 0) ws[SCALE_OFF + c] = (float)chunk_ex;

    // Gold-path partial score for this chunk's timesteps.
    float g = 0.0f;
    for (int tt = t0 + lane; tt < t1; tt += 32) {
        const int yt = y[tt];
        g += emit[x[tt] * TNUM + yt];
        g += (tt == 0) ? trans[yt]                        // trans[START=0, y0]
                       : trans[y[tt - 1] * TNUM + yt];
        if (tt == SEQL - 1) g += trans[yt * TNUM + 1];    // trans[yL, END=1]
    }
    g += SWZF(g, SWZ_X1);
    g += SWZF(g, SWZ_X2);
    g += SWZF(g, SWZ_X4);
    g += SWZF(g, SWZ_X8);
    g += SWZF(g, SWZ_X16);
    if (lane == 0) ws[GOLD_OFF + c] = g;
}

__global__ __launch_bounds__(32) void crf_phase2(
    const float* __restrict__ ws,
    const float* __restrict__ trans,
    float*       __restrict__ out)
{
    __shared__ float sv[16];
    const int lane = threadIdx.x & 31;
    const int j    = lane & 15;        // lanes 16-31 mirror lanes 0-15

    float  ve   = (j == 0) ? 1.0f : 0.0f;  // exp(init): 1 at START, 0 elsewhere
    double Lacc = 0.0;

    // Software-pipelined row loads: r[i] = Pe^T[j, i] of current chunk.
    float r[16];
    #pragma unroll
    for (int i = 0; i < 16; ++i) r[i] = ws[MAT_OFF + j * TNUM + i];

    for (int c = 0; c < NCHUNK; ++c) {
        if (lane < 16) sv[j] = ve;
        __syncthreads();
        float vv[16];
        #pragma unroll
        for (int i = 0; i < 16; ++i) vv[i] = sv[i];
        __syncthreads();

        float acc = 0.0f;
        #pragma unroll
        for (int i = 0; i < 16; ++i) acc = fmaf(r[i], vv[i], acc);

        if (c + 1 < NCHUNK) {
            #pragma unroll
            for (int i = 0; i < 16; ++i)
                r[i] = ws[MAT_OFF + (c + 1) * 256 + j * TNUM + i];
        }

        // Renormalize (max over the 16-lane half) and absorb logs in f64.
        float mx = acc;
        mx = fmaxf(mx, SWZF(mx, SWZ_X1));
        mx = fmaxf(mx, SWZF(mx, SWZ_X2));
        mx = fmaxf(mx, SWZF(mx, SWZ_X4));
        mx = fmaxf(mx, SWZF(mx, SWZ_X8));
        ve = acc / mx;
        const float sc = ws[SCALE_OFF + c];
        Lacc += (double)logf(mx) + (double)sc * 0.6931471805599453;
    }

    // z = Lacc + log( sum_j ve[j] * exp(trans[j, END]) )
    float term = ve * __expf(trans[j * TNUM + 1]);
    term += SWZF(term, SWZ_X1);
    term += SWZF(term, SWZ_X2);
    term += SWZF(term, SWZ_X4);
    term += SWZF(term, SWZ_X8);
    const double z = Lacc + (double)logf(term);

    // gold score = sum of per-chunk partials
    float g = 0.0f;
    for (int c = lane; c < NCHUNK; c += 32) g += ws[GOLD_OFF + c];
    g += SWZF(g, SWZ_X1);
    g += SWZF(g, SWZ_X2);
    g += SWZF(g, SWZ_X4);
    g += SWZF(g, SWZ_X8);
    g += SWZF(g, SWZ_X16);

    if (lane == 0) out[0] = (float)(z - (double)g);
}

extern "C" void kernel_launch(void* const* d_in, const int* in_sizes, int n_in,
                              void* d_out, int out_size, void* d_ws, size_t ws_size,
                              hipStream_t stream) {
    const float* emit  = (const float*)d_in[0];   // [50000, 16] f32
    const float* trans = (const float*)d_in[1];   // [16, 16] f32
    const int*   x     = (const int*)d_in[2];     // [8192] i32
    const int*   y     = (const int*)d_in[3];     // [8192] i32
    float*       ws    = (float*)d_ws;            // needs (128*256 + 256)*4 = ~132 KB
    float*       out   = (float*)d_out;           // scalar f32

    crf_phase1<<<NCHUNK, 32, 0, stream>>>(emit, trans, x, y, ws);
    crf_phase2<<<1, 32, 0, stream>>>(ws, trans, out);
}